// EfficientAttention3D_37735582662800
// MI455X (gfx1250) — compile-verified
//
#include <hip/hip_runtime.h>

typedef __attribute__((ext_vector_type(2))) float v2f;
typedef __attribute__((ext_vector_type(8))) float v8f;

#define C_IN   64
#define INNER  128
#define QKV_OUT 384
#define B_DIM  2
#define D_DIM  32
#define H_DIM  64
#define W_DIM  64
#define N_ROWS 4096   /* B*D*H */

/* workspace layout (float elements) */
#define KRAW_ELEMS (33554432ull)   /* 2*32*128*64*64 */
#define KCOL_ELEMS (524288ull)     /* 2*32*128*64    */
#define SQ_ELEMS   (1048576ull)    /* 4*4096*64      */
#define KMAX_OFF   (KRAW_ELEMS)
#define KINV_OFF   (KMAX_OFF + KCOL_ELEMS)
#define SQ_OFF     (KINV_OFF + KCOL_ELEMS)
#define SV_OFF     (SQ_OFF + SQ_ELEMS)

#define W_LD  68    /* wqkv LDS pitch (272B rows: 16B aligned, bank stride 4) */
#define X_LD  68    /* x LDS pitch    (16B-aligned rows for async B128)      */
#define WO_LD 132   /* wout LDS pitch (528B rows: 16B aligned, bank stride 4)*/
#define OM_LD 66    /* outmat LDS pitch (ds-only, keep 2-float skew)         */

// Async global->LDS copy of one 16B chunk per active lane (GVS addressing).
// Tracked on ASYNCcnt; wave later issues s_wait_asynccnt 0 before the barrier.
__device__ __forceinline__ void async_copy_b128(uint32_t lds_addr,
                                                const void* gbase,
                                                uint32_t goff) {
  asm volatile("global_load_async_to_lds_b128 %0, %1, %2"
               :
               : "v"(lds_addr), "v"(goff), "s"(gbase)
               : "memory");
}
__device__ __forceinline__ void async_wait0() {
  asm volatile("s_wait_asynccnt 0" ::: "memory");
}

// -------------------------------------------------------------------------
// K1: qkv GEMM (W_qkv 384x64 @ x 64x64) per (b,d,h) row, fused epilogues:
//   q waves  -> softmax over W, write Sq[head,row,w]
//   v waves  -> channel sum,    write Sv[head,row,w]
//   k waves  -> store raw k to workspace
// -------------------------------------------------------------------------
__global__ __launch_bounds__(384) void ea3d_qkv_kernel(
    const float* __restrict__ x, const float* __restrict__ wqkv,
    float* __restrict__ ws) {
  __shared__ __align__(16) float wlds[QKV_OUT * W_LD];
  __shared__ __align__(16) float xlds[C_IN * X_LD];

  const int tid = threadIdx.x;
  const int bid = blockIdx.x;           // row id = bd*64 + h
  const int bd = bid >> 6, h = bid & 63;
  const int b = bd >> 5, d = bd & 31;

  // stage W_qkv (384x64) via async B128: 16 chunks per row, 6144 chunks
  for (int idx = tid; idx < QKV_OUT * 16; idx += 384) {
    int row = idx >> 4, ch = idx & 15;
    async_copy_b128((uint32_t)(size_t)&wlds[row * W_LD + ch * 4], wqkv,
                    (uint32_t)((row * 64 + ch * 4) * sizeof(float)));
  }
  // stage x row-slab (64 ch x 64 w) via async B128: 1024 chunks
  const float* xrow = x + ((((size_t)b * C_IN) * D_DIM + d) * H_DIM + h) * W_DIM;
  for (int idx = tid; idx < C_IN * 16; idx += 384) {
    int c = idx >> 4, ch = idx & 15;
    async_copy_b128((uint32_t)(size_t)&xlds[c * X_LD + ch * 4], xrow,
                    (uint32_t)(((uint32_t)c * (D_DIM * H_DIM * W_DIM) + ch * 4) *
                               sizeof(float)));
  }
  async_wait0();
  __syncthreads();

  const int wv = tid >> 5;
  const int lane = tid & 31;
  const int hf = lane >> 4;      // which 16-lane half
  const int l  = lane & 15;

  int chanBase, region, head;
  if (wv < 4)      { region = 0; head = wv;     chanBase = wv * 32; }             // q
  else if (wv < 8) { region = 2; head = wv - 4; chanBase = 256 + (wv - 4) * 32; } // v
  else             { region = 1; head = 0;      chanBase = 128 + (wv - 8) * 32; } // k

  v8f acc[2][4];
#pragma unroll
  for (int mt = 0; mt < 2; ++mt)
#pragma unroll
    for (int nt = 0; nt < 4; ++nt) acc[mt][nt] = (v8f){};

  for (int kk = 0; kk < 16; ++kk) {
    const int k0 = kk * 4 + hf * 2;
    v2f bf[4];
#pragma unroll
    for (int nt = 0; nt < 4; ++nt) {
      bf[nt].x = xlds[k0 * X_LD + nt * 16 + l];
      bf[nt].y = xlds[(k0 + 1) * X_LD + nt * 16 + l];
    }
#pragma unroll
    for (int mt = 0; mt < 2; ++mt) {
      v2f af;
      af.x = wlds[(chanBase + mt * 16 + l) * W_LD + k0];
      af.y = wlds[(chanBase + mt * 16 + l) * W_LD + k0 + 1];
#pragma unroll
      for (int nt = 0; nt < 4; ++nt)
        acc[mt][nt] = __builtin_amdgcn_wmma_f32_16x16x4_f32(
            false, af, false, bf[nt], (short)0, acc[mt][nt], false, false);
    }
  }

  float* kraw = ws;
  float* Sq = ws + SQ_OFF;
  float* Sv = ws + SV_OFF;

  if (region == 1) {
    // store raw k: layout [bd][c(128)][h][w]
#pragma unroll
    for (int mt = 0; mt < 2; ++mt)
#pragma unroll
      for (int r = 0; r < 8; ++r) {
        int c = (chanBase - 128) + mt * 16 + r + 8 * hf;
        size_t base = (((size_t)bd * 128 + c) * 64 + h) * 64;
#pragma unroll
        for (int nt = 0; nt < 4; ++nt)
          kraw[base + nt * 16 + l] = acc[mt][nt][r];
      }
  } else {
    if (region == 0) {
      // softmax over W per channel (values live across lanes of the half + nt)
#pragma unroll
      for (int mt = 0; mt < 2; ++mt)
#pragma unroll
        for (int r = 0; r < 8; ++r) {
          float m = -3.4e38f;
#pragma unroll
          for (int nt = 0; nt < 4; ++nt) m = fmaxf(m, acc[mt][nt][r]);
#pragma unroll
          for (int dd = 1; dd < 16; dd <<= 1) m = fmaxf(m, __shfl_xor(m, dd, 32));
          float s = 0.f;
#pragma unroll
          for (int nt = 0; nt < 4; ++nt) {
            float e = __expf(acc[mt][nt][r] - m);
            acc[mt][nt][r] = e;
            s += e;
          }
#pragma unroll
          for (int dd = 1; dd < 16; dd <<= 1) s += __shfl_xor(s, dd, 32);
          float inv = 1.f / s;
#pragma unroll
          for (int nt = 0; nt < 4; ++nt) acc[mt][nt][r] *= inv;
        }
    }
    // Sq / Sv: sum over the head's 32 channels at each voxel
    float* dst = (region == 0) ? Sq : Sv;
#pragma unroll
    for (int nt = 0; nt < 4; ++nt) {
      float t = 0.f;
#pragma unroll
      for (int mt = 0; mt < 2; ++mt)
#pragma unroll
        for (int r = 0; r < 8; ++r) t += acc[mt][nt][r];
      t += __shfl_xor(t, 16, 32);
      if (lane < 16)
        dst[((size_t)head * N_ROWS + bid) * 64 + nt * 16 + l] = t;
    }
  }
}

// -------------------------------------------------------------------------
// K2: per k-column (b,d,c,w): max / 1/sum-exp over H (softmax stats)
// -------------------------------------------------------------------------
__global__ __launch_bounds__(256) void ea3d_kstats_kernel(float* __restrict__ ws) {
  const float* kraw = ws;
  float* kmax = ws + KMAX_OFF;
  float* kinv = ws + KINV_OFF;
  const int col = blockIdx.x * 256 + threadIdx.x;   // 524288 columns
  const int w = col & 63;
  const int c = (col >> 6) & 127;
  const int bd = col >> 13;
  const size_t base = (((size_t)bd * 128 + c) * 64) * 64 + w;
  float vals[64];
  float m = -3.4e38f;
#pragma unroll
  for (int hh = 0; hh < 64; ++hh) {
    vals[hh] = kraw[base + (size_t)hh * 64];
    m = fmaxf(m, vals[hh]);
  }
  float s = 0.f;
#pragma unroll
  for (int hh = 0; hh < 64; ++hh) s += __expf(vals[hh] - m);
  kmax[col] = m;
  kinv[col] = 1.f / s;
}

// -------------------------------------------------------------------------
// K3: out = k_sm * Sq * Sv  (built in LDS), then y = W_out @ out + b + x
// -------------------------------------------------------------------------
__global__ __launch_bounds__(256) void ea3d_out_kernel(
    const float* __restrict__ x, const float* __restrict__ wout,
    const float* __restrict__ bout, const float* __restrict__ ws,
    float* __restrict__ y) {
  __shared__ __align__(16) float wo[64 * WO_LD];
  __shared__ __align__(16) float om[128 * OM_LD];
  __shared__ __align__(16) float sl[4 * 64];

  const int tid = threadIdx.x;
  const int bid = blockIdx.x;
  const int bd = bid >> 6, h = bid & 63;
  const int b = bd >> 5, d = bd & 31;

  const float* kraw = ws;
  const float* kmax = ws + KMAX_OFF;
  const float* kinv = ws + KINV_OFF;
  const float* Sq = ws + SQ_OFF;
  const float* Sv = ws + SV_OFF;

  // stage W_out (64x128) via async B128: 32 chunks per row, 2048 chunks
  for (int idx = tid; idx < 64 * 32; idx += 256) {
    int row = idx >> 5, ch = idx & 31;
    async_copy_b128((uint32_t)(size_t)&wo[row * WO_LD + ch * 4], wout,
                    (uint32_t)((row * 128 + ch * 4) * sizeof(float)));
  }
  {
    int head = tid >> 6, w = tid & 63;
    size_t sidx = ((size_t)head * N_ROWS + bid) * 64 + w;
    sl[tid] = Sq[sidx] * Sv[sidx];
  }
  async_wait0();
  __syncthreads();

  for (int idx = tid; idx < 128 * 64; idx += 256) {
    int hc = idx >> 6, w = idx & 63;
    size_t kcol = ((size_t)bd * 128 + hc) * 64 + w;
    float kv = kraw[(((size_t)bd * 128 + hc) * 64 + h) * 64 + w];
    float e = __expf(kv - kmax[kcol]) * kinv[kcol];
    om[hc * OM_LD + w] = e * sl[(hc >> 5) * 64 + w];
  }
  __syncthreads();

  const int wv = tid >> 5;
  const int lane = tid & 31;
  const int hf = lane >> 4;
  const int l = lane & 15;
  const int mt = wv >> 1;            // M tile (0..3)
  const int ntB = (wv & 1) * 2;      // N tiles {ntB, ntB+1}

  v8f acc[2];
  acc[0] = (v8f){};
  acc[1] = (v8f){};

  for (int kk = 0; kk < 32; ++kk) {
    const int k0 = kk * 4 + hf * 2;
    v2f af;
    af.x = wo[(mt * 16 + l) * WO_LD + k0];
    af.y = wo[(mt * 16 + l) * WO_LD + k0 + 1];
#pragma unroll
    for (int j = 0; j < 2; ++j) {
      v2f bf;
      bf.x = om[k0 * OM_LD + (ntB + j) * 16 + l];
      bf.y = om[(k0 + 1) * OM_LD + (ntB + j) * 16 + l];
      acc[j] = __builtin_amdgcn_wmma_f32_16x16x4_f32(
          false, af, false, bf, (short)0, acc[j], false, false);
    }
  }

#pragma unroll
  for (int j = 0; j < 2; ++j) {
    int nt = ntB + j;
#pragma unroll
    for (int r = 0; r < 8; ++r) {
      int o = mt * 16 + r + 8 * hf;
      int w = nt * 16 + l;
      size_t xi = ((((size_t)b * 64 + o) * 32 + d) * 64 + h) * 64 + w;
      y[xi] = acc[j][r] + bout[o] + x[xi];
    }
  }
}

extern "C" void kernel_launch(void* const* d_in, const int* in_sizes, int n_in,
                              void* d_out, int out_size, void* d_ws, size_t ws_size,
                              hipStream_t stream) {
  (void)in_sizes; (void)n_in; (void)out_size; (void)ws_size;
  const float* x    = (const float*)d_in[0];
  const float* wqkv = (const float*)d_in[1];
  const float* wout = (const float*)d_in[2];
  const float* bout = (const float*)d_in[3];
  float* ws = (float*)d_ws;
  float* y  = (float*)d_out;

  ea3d_qkv_kernel<<<N_ROWS, 384, 0, stream>>>(x, wqkv, ws);
  ea3d_kstats_kernel<<<KCOL_ELEMS / 256, 256, 0, stream>>>(ws);
  ea3d_out_kernel<<<N_ROWS, 256, 0, stream>>>(x, wout, bout, ws, y);
}